// DSC_AttentionBaseline_75960791597259
// MI455X (gfx1250) — compile-verified
//
#include <hip/hip_runtime.h>
#include <hip/hip_bf16.h>

// ---------------------------------------------------------------------------
// DSC attention baseline for MI455X (gfx1250, wave32).
// All contractions on V_WMMA_F32_16X16X4_F32 (fp32 matrix pipe, matches fp32
// reference numerics). One wave owns a 32x64 (GEMM) or 32x32 (conv) C strip:
// two A fragments and four (two) B fragments give 8 (4) WMMAs per K-step with
// every B fragment reused across two M-tiles -> 1.5 loads per WMMA.
// Strides are template constants so all secondary loads use immediate
// instruction offsets; only 2 pointer increments per K-step.
// Workspace: ~165 MB carved from d_ws.
// ---------------------------------------------------------------------------

typedef __attribute__((ext_vector_type(2))) float f32x2;
typedef __attribute__((ext_vector_type(8))) float f32x8;

#define NCH   256
#define KTAP  9
#define SCALE 0.17677669529663687f

__device__ __forceinline__ f32x8 wmma4(f32x2 a, f32x2 b, f32x8 c) {
  return __builtin_amdgcn_wmma_f32_16x16x4_f32(false, a, false, b,
                                               (short)0, c, false, false);
}

// ---------------------------------------------------------------------------
// Strided batched GEMM: C[b,m,n] = alpha * sum_k A(m,k)*B(k,n) [+ bias]
// Wave computes a 32(M) x 64(N) strip: 2 A fragments x 4 B fragments.
// Requires: Mdim % 32 == 0 (or ==32), Ndim % 256 == 0, Kdim % 8 == 0.
// grid: (N/256, M/32, batch), block 128 (4 waves)
// ---------------------------------------------------------------------------
template<long SAM, long SAK, long SBK, long SBN, long SCM, long SCN, int BIAS>
__global__ void gemm_wmma_f32(const float* __restrict__ A,
                              const float* __restrict__ B,
                              float* __restrict__ C,
                              int Kdim, long sAb, long sBb, long sCb,
                              const float* __restrict__ bias, float alpha)
{
  const int lane = threadIdx.x & 31;
  const int wave = threadIdx.x >> 5;
  const int h = lane >> 4;          // lane-half -> K pair / row half
  const int l = lane & 15;
  const int m0 = blockIdx.y * 32;
  const int n0 = (blockIdx.x * 4 + wave) * 64;
  const int bz = blockIdx.z;

  const float* pA = A + (long)bz * sAb + (long)(m0 + l) * SAM + (long)(2 * h) * SAK;
  const float* pB = B + (long)bz * sBb + (long)(n0 + l) * SBN + (long)(2 * h) * SBK;

  f32x8 ac00 = {}, ac01 = {}, ac02 = {}, ac03 = {};
  f32x8 ac10 = {}, ac11 = {}, ac12 = {}, ac13 = {};
#pragma unroll 2
  for (int kk = 0; kk < Kdim; kk += 4) {
    f32x2 aL; aL.x = pA[0];                  aL.y = pA[SAK];
    f32x2 aH; aH.x = pA[16 * SAM];           aH.y = pA[16 * SAM + SAK];
    f32x2 b0; b0.x = pB[0];                  b0.y = pB[SBK];
    f32x2 b1; b1.x = pB[16 * SBN];           b1.y = pB[16 * SBN + SBK];
    f32x2 b2; b2.x = pB[32 * SBN];           b2.y = pB[32 * SBN + SBK];
    f32x2 b3; b3.x = pB[48 * SBN];           b3.y = pB[48 * SBN + SBK];
    __builtin_prefetch(pA + 16 * SAK, 0, 3);
    __builtin_prefetch(pB + 16 * SBK, 0, 3);
    ac00 = wmma4(aL, b0, ac00);
    ac10 = wmma4(aH, b0, ac10);
    ac01 = wmma4(aL, b1, ac01);
    ac11 = wmma4(aH, b1, ac11);
    ac02 = wmma4(aL, b2, ac02);
    ac12 = wmma4(aH, b2, ac12);
    ac03 = wmma4(aL, b3, ac03);
    ac13 = wmma4(aH, b3, ac13);
    pA += 4 * SAK;
    pB += 4 * SBK;
  }

  float* Cb = C + (long)bz * sCb;
#pragma unroll
  for (int g = 0; g < 2; ++g) {
#pragma unroll
    for (int u = 0; u < 4; ++u) {
      const f32x8 acc =
          (g == 0) ? ((u == 0) ? ac00 : (u == 1) ? ac01 : (u == 2) ? ac02 : ac03)
                   : ((u == 0) ? ac10 : (u == 1) ? ac11 : (u == 2) ? ac12 : ac13);
      const int col = n0 + 16 * u + l;
      float badd = 0.f;
      if (BIAS == 2) badd = bias[col];
#pragma unroll
      for (int v = 0; v < 8; ++v) {
        const int row = m0 + 16 * g + v + 8 * h;
        float val = acc[v] * alpha + badd;
        if (BIAS == 1) val += bias[row];
        Cb[(long)row * SCM + (long)col * SCN] = val;
      }
    }
  }
}

// ---------------------------------------------------------------------------
// Implicit-GEMM 3x3 conv, pad 1, stride 1, 32x32 images; K = Cin*9 (constant).
// Wave computes 32(M) x 32(N): 2 A fragments x 2 gathered B fragments.
// Zero padding via clamped address + 0/1 mask (no exec divergence).
// A-row guard via clamp+mask (handles Cout=18 with a single 32-row tile).
// grid: (8, ceil(Cout/32), batch), block 128
// ---------------------------------------------------------------------------
template<int CIN, int COUT>
__global__ void conv3x3_wmma(const float* __restrict__ X,
                             const float* __restrict__ Wt,
                             const float* __restrict__ bias,
                             float* __restrict__ Out,
                             long bsX, long bsOut)
{
  const int lane = threadIdx.x & 31;
  const int wave = threadIdx.x >> 5;
  const int h = lane >> 4, l = lane & 15;
  const int n0 = (blockIdx.x * 4 + wave) * 32;
  const int m0 = blockIdx.y * 32;
  const int bz = blockIdx.z;
  const int Kdim = CIN * 9;
  const int q0 = n0 + l, q1 = n0 + 16 + l;
  const int oy0 = q0 >> 5, ox0 = q0 & 31;
  const int oy1 = q1 >> 5, ox1 = q1 & 31;
  const int mLo = m0 + l, mHi = m0 + 16 + l;
  const float mskLo = (mLo < COUT) ? 1.f : 0.f;
  const float mskHi = (mHi < COUT) ? 1.f : 0.f;
  const int rowLo = min(mLo, COUT - 1);
  const int rowHi = min(mHi, COUT - 1);
  const float* Xb = X + (long)bz * bsX;
  const float* pWlo = Wt + (long)rowLo * Kdim + 2 * h;
  const float* pWhi = Wt + (long)rowHi * Kdim + 2 * h;

  int tL = 2 * h;   // k2 = cL*9 + tL for this lane's first element
  int cL = 0;

  f32x8 ac00 = {}, ac01 = {}, ac10 = {}, ac11 = {};
  for (int kk = 0; kk < Kdim; kk += 4) {
    f32x2 aL; aL.x = pWlo[0] * mskLo; aL.y = pWlo[1] * mskLo;
    f32x2 aH; aH.x = pWhi[0] * mskHi; aH.y = pWhi[1] * mskHi;
    float bv0[2], bv1[2];
#pragma unroll
    for (int j = 0; j < 2; ++j) {
      int t = tL + j, c = cL;
      if (t >= 9) { t -= 9; ++c; }
      const int dy = t / 3 - 1;
      const int dx = t - (t / 3) * 3 - 1;
      const float* Xc = Xb + ((long)c << 10);
      {
        const int yy = oy0 + dy, xx = ox0 + dx;
        const int yc = min(max(yy, 0), 31), xc = min(max(xx, 0), 31);
        const float m = (yy == yc && xx == xc) ? 1.f : 0.f;
        bv0[j] = Xc[(yc << 5) + xc] * m;
      }
      {
        const int yy = oy1 + dy, xx = ox1 + dx;
        const int yc = min(max(yy, 0), 31), xc = min(max(xx, 0), 31);
        const float m = (yy == yc && xx == xc) ? 1.f : 0.f;
        bv1[j] = Xc[(yc << 5) + xc] * m;
      }
    }
    f32x2 b0; b0.x = bv0[0]; b0.y = bv0[1];
    f32x2 b1; b1.x = bv1[0]; b1.y = bv1[1];
    ac00 = wmma4(aL, b0, ac00);
    ac10 = wmma4(aH, b0, ac10);
    ac01 = wmma4(aL, b1, ac01);
    ac11 = wmma4(aH, b1, ac11);
    pWlo += 4; pWhi += 4;
    tL += 4; if (tL >= 9) { tL -= 9; ++cL; }
  }

  float* Ob = Out + (long)bz * bsOut;
#pragma unroll
  for (int v = 0; v < 8; ++v) {
    const int rLo = m0 + v + 8 * h;
    const int rHi = m0 + 16 + v + 8 * h;
    if (rLo < COUT) {
      Ob[(long)rLo * 1024 + q0] = ac00[v] + bias[rLo];
      Ob[(long)rLo * 1024 + q1] = ac01[v] + bias[rLo];
    }
    if (rHi < COUT) {
      Ob[(long)rHi * 1024 + q0] = ac10[v] + bias[rHi];
      Ob[(long)rHi * 1024 + q1] = ac11[v] + bias[rHi];
    }
  }
}

// ---------------------------------------------------------------------------
// Deformable conv (implicit GEMM, K = 256*9) with fused bilinear sampling.
// morph0: tap t -> sample (oh*9+t)*32+ow ; morph1: tap t -> oh*288+ow*9+t
// Wave computes 32 x 32 (4 accumulators, bilinear B reused across M-tiles).
// grid: (8, 8, 16), block 128
// ---------------------------------------------------------------------------
__device__ __forceinline__ float bil_sample(const float* __restrict__ Xc,
                                            float y, float x)
{
  const float y0f = floorf(y), x0f = floorf(x);
  const int y0 = (int)y0f, x0 = (int)x0f;
  const int y1 = min(y0 + 1, 31), x1 = min(x0 + 1, 31);
  const float wy = y - y0f, wx = x - x0f;
  const float v00 = Xc[(y0 << 5) + x0];
  const float v01 = Xc[(y0 << 5) + x1];
  const float v10 = Xc[(y1 << 5) + x0];
  const float v11 = Xc[(y1 << 5) + x1];
  return (1.f - wy) * ((1.f - wx) * v00 + wx * v01)
       +        wy  * ((1.f - wx) * v10 + wx * v11);
}

template<int MORPH>
__global__ void dsconv_wmma(const float* __restrict__ X,
                            const float* __restrict__ Wt,
                            const float* __restrict__ bias,
                            const float* __restrict__ sampY,
                            const float* __restrict__ sampX,
                            float* __restrict__ Out,
                            long bsOut)
{
  const int lane = threadIdx.x & 31;
  const int wave = threadIdx.x >> 5;
  const int h = lane >> 4, l = lane & 15;
  const int n0 = (blockIdx.x * 4 + wave) * 32;
  const int m0 = blockIdx.y * 32;
  const int bz = blockIdx.z;
  const int Kdim = NCH * 9;       // 2304
  const int q0 = n0 + l, q1 = n0 + 16 + l;
  const int oh0 = q0 >> 5, ow0 = q0 & 31;
  const int oh1 = q1 >> 5, ow1 = q1 & 31;
  const float* Xb = X + (long)bz * (NCH * 1024);
  const float* sY = sampY + (long)bz * (KTAP * 1024);
  const float* sX = sampX + (long)bz * (KTAP * 1024);
  const float* pW = Wt + (long)(m0 + l) * Kdim + 2 * h;   // hi rows at +16*Kdim

  int tL = 2 * h;
  int cL = 0;

  f32x8 ac00 = {}, ac01 = {}, ac10 = {}, ac11 = {};
  for (int kk = 0; kk < Kdim; kk += 4) {
    f32x2 aL; aL.x = pW[0];             aL.y = pW[1];
    f32x2 aH; aH.x = pW[16 * 2304];     aH.y = pW[16 * 2304 + 1];
    float bv0[2], bv1[2];
#pragma unroll
    for (int j = 0; j < 2; ++j) {
      int t = tL + j, c = cL;
      if (t >= 9) { t -= 9; ++c; }
      const float* Xc = Xb + ((long)c << 10);
      const int s0 = (MORPH == 0) ? ((oh0 * 9 + t) * 32 + ow0)
                                  : (oh0 * 288 + ow0 * 9 + t);
      const int s1 = (MORPH == 0) ? ((oh1 * 9 + t) * 32 + ow1)
                                  : (oh1 * 288 + ow1 * 9 + t);
      bv0[j] = bil_sample(Xc, sY[s0], sX[s0]);
      bv1[j] = bil_sample(Xc, sY[s1], sX[s1]);
    }
    f32x2 b0; b0.x = bv0[0]; b0.y = bv0[1];
    f32x2 b1; b1.x = bv1[0]; b1.y = bv1[1];
    ac00 = wmma4(aL, b0, ac00);
    ac10 = wmma4(aH, b0, ac10);
    ac01 = wmma4(aL, b1, ac01);
    ac11 = wmma4(aH, b1, ac11);
    pW += 4;
    tL += 4; if (tL >= 9) { tL -= 9; ++cL; }
  }

  float* Ob = Out + (long)bz * bsOut;
#pragma unroll
  for (int v = 0; v < 8; ++v) {
    const int rLo = m0 + v + 8 * h;
    const int rHi = m0 + 16 + v + 8 * h;
    Ob[(long)rLo * 1024 + q0] = ac00[v] + bias[rLo];
    Ob[(long)rLo * 1024 + q1] = ac01[v] + bias[rLo];
    Ob[(long)rHi * 1024 + q0] = ac10[v] + bias[rHi];
    Ob[(long)rHi * 1024 + q1] = ac11[v] + bias[rHi];
  }
}

// ---------------------------------------------------------------------------
// GroupNorm (+ relu(1) / tanh(2)), in place. grid: (groups, batch), block 256
// ---------------------------------------------------------------------------
__global__ void groupnorm_act(float* __restrict__ x,
                              const float* __restrict__ gamma,
                              const float* __restrict__ beta,
                              int cPerG, long batchStride, int act)
{
  const int grp = blockIdx.x, bz = blockIdx.y;
  const long base = (long)bz * batchStride + (long)grp * cPerG * 1024;
  const int N = cPerG * 1024;
  __shared__ float r0[256], r1[256];
  float s = 0.f, ss = 0.f;
  for (int i = threadIdx.x; i < N; i += 256) {
    float v = x[base + i];
    s += v; ss += v * v;
  }
  r0[threadIdx.x] = s; r1[threadIdx.x] = ss;
  __syncthreads();
  for (int st = 128; st > 0; st >>= 1) {
    if (threadIdx.x < st) {
      r0[threadIdx.x] += r0[threadIdx.x + st];
      r1[threadIdx.x] += r1[threadIdx.x + st];
    }
    __syncthreads();
  }
  const float mu = r0[0] / (float)N;
  const float var = r1[0] / (float)N - mu * mu;
  const float inv = rsqrtf(var + 1e-5f);
  for (int i = threadIdx.x; i < N; i += 256) {
    const int c = grp * cPerG + (i >> 10);
    float v = (x[base + i] - mu) * inv * gamma[c] + beta[c];
    if (act == 1) v = fmaxf(v, 0.f);
    else if (act == 2) v = tanhf(v);
    x[base + i] = v;
  }
}

// ---------------------------------------------------------------------------
// Snake cumsum -> clipped sampling coordinates. grid 64 x 256 threads.
// ---------------------------------------------------------------------------
__global__ void coords_kernel(const float* __restrict__ off,
                              float* __restrict__ sampY,
                              float* __restrict__ sampX, int morph)
{
  const int idx = blockIdx.x * 256 + threadIdx.x;
  const int bz = idx >> 10;
  const int p = idx & 1023;
  const int hh = p >> 5, ww = p & 31;
  const long obase = (long)bz * (18 * 1024) + (morph ? 9 * 1024 : 0);
  float o[9];
#pragma unroll
  for (int k = 0; k < 9; ++k) o[k] = off[obase + (long)k * 1024 + p];
  float s[9];
  s[4] = 0.f;
  s[3] = o[3]; s[2] = s[3] + o[2]; s[1] = s[2] + o[1]; s[0] = s[1] + o[0];
  s[5] = o[5]; s[6] = s[5] + o[6]; s[7] = s[6] + o[7]; s[8] = s[7] + o[8];
  const long sbase = (long)bz * (KTAP * 1024);
#pragma unroll
  for (int k = 0; k < 9; ++k) {
    float y, x; int si;
    if (morph == 0) {
      y = (float)hh + s[k];
      x = (float)(ww + (k - 4));
      si = (hh * 9 + k) * 32 + ww;
    } else {
      x = (float)ww + s[k];
      y = (float)(hh + (k - 4));
      si = hh * 288 + ww * 9 + k;
    }
    sampY[sbase + si] = fminf(fmaxf(y, 0.f), 31.f);
    sampX[sbase + si] = fminf(fmaxf(x, 0.f), 31.f);
  }
}

// ---------------------------------------------------------------------------
// attn[b,i,j] = fm[b,j,i] + fm[b,i,j]; softmax over j. grid (1024,16), 256thr
// ---------------------------------------------------------------------------
__global__ void transadd_softmax(const float* __restrict__ fm,
                                 float* __restrict__ attn)
{
  const int i = blockIdx.x, bz = blockIdx.y;
  const float* base = fm + (long)bz * 1048576;
  float* out = attn + (long)bz * 1048576 + (long)i * 1024;
  __shared__ float buf[1024];
  __shared__ float red[256];
  float mx = -3.402823466e38f;
  for (int n = threadIdx.x; n < 1024; n += 256) {
    float v = base[(long)i * 1024 + n] + base[(long)n * 1024 + i];
    buf[n] = v;
    mx = fmaxf(mx, v);
  }
  red[threadIdx.x] = mx; __syncthreads();
  for (int st = 128; st > 0; st >>= 1) {
    if (threadIdx.x < st) red[threadIdx.x] = fmaxf(red[threadIdx.x], red[threadIdx.x + st]);
    __syncthreads();
  }
  mx = red[0]; __syncthreads();
  float sum = 0.f;
  for (int n = threadIdx.x; n < 1024; n += 256) {
    float e = __expf(buf[n] - mx);
    buf[n] = e; sum += e;
  }
  red[threadIdx.x] = sum; __syncthreads();
  for (int st = 128; st > 0; st >>= 1) {
    if (threadIdx.x < st) red[threadIdx.x] += red[threadIdx.x + st];
    __syncthreads();
  }
  const float inv = 1.f / red[0];
  for (int n = threadIdx.x; n < 1024; n += 256) out[n] = buf[n] * inv;
}

// ---------------------------------------------------------------------------
extern "C" void kernel_launch(void* const* d_in, const int* in_sizes, int n_in,
                              void* d_out, int out_size, void* d_ws, size_t ws_size,
                              hipStream_t stream) {
  const float* x     = (const float*)d_in[0];
  const float* wq    = (const float*)d_in[1];
  const float* wk    = (const float*)d_in[2];
  const float* wv    = (const float*)d_in[3];
  const float* wo    = (const float*)d_in[4];
  const float* bq    = (const float*)d_in[5];
  const float* bk    = (const float*)d_in[6];
  const float* bv    = (const float*)d_in[7];
  const float* bo    = (const float*)d_in[8];
  const float* p1_w  = (const float*)d_in[9];
  const float* p1_b  = (const float*)d_in[10];
  const float* p2_w  = (const float*)d_in[11];
  const float* p2_b  = (const float*)d_in[12];
  const float* c0_w  = (const float*)d_in[13];
  const float* c0_b  = (const float*)d_in[14];
  const float* c0_g  = (const float*)d_in[15];
  const float* c0_be = (const float*)d_in[16];
  const float* cx_off_w = (const float*)d_in[17];
  const float* cx_off_b = (const float*)d_in[18];
  const float* cx_gno_g = (const float*)d_in[19];
  const float* cx_gno_b = (const float*)d_in[20];
  const float* cx_dsc_w = (const float*)d_in[21];
  const float* cx_dsc_b = (const float*)d_in[22];
  const float* cx_gn_g  = (const float*)d_in[23];
  const float* cx_gn_b  = (const float*)d_in[24];
  const float* cy_off_w = (const float*)d_in[25];
  const float* cy_off_b = (const float*)d_in[26];
  const float* cy_gno_g = (const float*)d_in[27];
  const float* cy_gno_b = (const float*)d_in[28];
  const float* cy_dsc_w = (const float*)d_in[29];
  const float* cy_dsc_b = (const float*)d_in[30];
  const float* cy_gn_g  = (const float*)d_in[31];
  const float* cy_gn_b  = (const float*)d_in[32];
  const float* c1_w  = (const float*)d_in[33];
  const float* c1_b  = (const float*)d_in[34];
  const float* c1_g  = (const float*)d_in[35];
  const float* c1_be = (const float*)d_in[36];
  float* out = (float*)d_out;

  // Workspace layout (floats). fmp2 aliases fmN+fmcat (both dead by then).
  float* ws = (float*)d_ws;
  float* qbuf  = ws;                       // 524288
  float* kbuf  = ws + 524288;              // 524288
  float* vbuf  = ws + 1048576;             // 524288
  float* attn  = ws + 1572864;             // 16777216
  float* fmN   = ws + 18350080;            // 4194304  (b,c,n) NCHW fm
  float* fmcat = ws + 22544384;            // 12582912 (b, 768, 1024)
  float* fmp2  = ws + 18350080;            // 16777216 (aliases fmN+fmcat)
  float* offX  = ws + 35127296;            // 294912
  float* offY  = ws + 35422208;            // 294912
  float* sY0   = ws + 35717120;            // 147456
  float* sX0   = ws + 35864576;            // 147456
  float* sY1   = ws + 36012032;            // 147456
  float* sX1   = ws + 36159488;            // 147456
  float* fmc1  = ws + 36306944;            // 4194304
  float* outb  = ws + 40501248;            // 524288
  (void)ws_size; (void)n_in; (void)in_sizes; (void)out_size;

  const dim3 blk(128);

  // q,k,v projections: y[b,o,p] = W[o,c] x[b,c,p] + bias[o]   (batch = 2)
  gemm_wmma_f32<256, 1, 1024, 1, 1024, 1, 1><<<dim3(4, 8, 2), blk, 0, stream>>>(
      wq, x, qbuf, 256, 0, 262144, 262144, bq, 1.f);
  gemm_wmma_f32<256, 1, 1024, 1, 1024, 1, 1><<<dim3(4, 8, 2), blk, 0, stream>>>(
      wk, x, kbuf, 256, 0, 262144, 262144, bk, 1.f);
  gemm_wmma_f32<256, 1, 1024, 1, 1024, 1, 1><<<dim3(4, 8, 2), blk, 0, stream>>>(
      wv, x, vbuf, 256, 0, 262144, 262144, bv, 1.f);

  // attn1[bh,m,n] = SCALE * sum_hc q[bh,hc,m] k[bh,hc,n]   (batch 16, K=32)
  gemm_wmma_f32<1, 1024, 1024, 1, 1024, 1, 0><<<dim3(4, 32, 16), blk, 0, stream>>>(
      qbuf, kbuf, attn, 32, 32768, 32768, 1048576, nullptr, SCALE);

  // fmN[b,c,n] = sum_m attn[b,m,n] p1_w[c,m] + p1_b[c]   (M=n, N=c, K=m)
  gemm_wmma_f32<1, 1024, 1, 1024, 1, 1024, 2><<<dim3(1, 32, 16), blk, 0, stream>>>(
      attn, p1_w, fmN, 1024, 1048576, 0, 262144, p1_b, 1.f);

  // fm1 = relu(GN(conv3x3(fmN)))  -> fmcat[:, 0:256]
  conv3x3_wmma<256, 256><<<dim3(8, 8, 16), blk, 0, stream>>>(
      fmN, c0_w, c0_b, fmcat, 262144, 786432);
  groupnorm_act<<<dim3(64, 16), dim3(256), 0, stream>>>(fmcat, c0_g, c0_be,
      4, 786432, 1);

  // ---- deformable branch cx (morph 0) ----
  conv3x3_wmma<256, 18><<<dim3(8, 1, 16), blk, 0, stream>>>(
      fmN, cx_off_w, cx_off_b, offX, 262144, 18432);
  groupnorm_act<<<dim3(9, 16), dim3(256), 0, stream>>>(offX, cx_gno_g, cx_gno_b,
      2, 18432, 2);
  coords_kernel<<<dim3(64), dim3(256), 0, stream>>>(offX, sY0, sX0, 0);
  dsconv_wmma<0><<<dim3(8, 8, 16), blk, 0, stream>>>(
      fmN, cx_dsc_w, cx_dsc_b, sY0, sX0, fmcat + 262144, 786432);
  groupnorm_act<<<dim3(64, 16), dim3(256), 0, stream>>>(fmcat + 262144,
      cx_gn_g, cx_gn_b, 4, 786432, 1);

  // ---- deformable branch cy (morph 1) ----
  conv3x3_wmma<256, 18><<<dim3(8, 1, 16), blk, 0, stream>>>(
      fmN, cy_off_w, cy_off_b, offY, 262144, 18432);
  groupnorm_act<<<dim3(9, 16), dim3(256), 0, stream>>>(offY, cy_gno_g, cy_gno_b,
      2, 18432, 2);
  coords_kernel<<<dim3(64), dim3(256), 0, stream>>>(offY, sY1, sX1, 1);
  dsconv_wmma<1><<<dim3(8, 8, 16), blk, 0, stream>>>(
      fmN, cy_dsc_w, cy_dsc_b, sY1, sX1, fmcat + 524288, 786432);
  groupnorm_act<<<dim3(64, 16), dim3(256), 0, stream>>>(fmcat + 524288,
      cy_gn_g, cy_gn_b, 4, 786432, 1);

  // c1: relu(GN(conv3x3(concat)))  (Cin = 768)
  conv3x3_wmma<768, 256><<<dim3(8, 8, 16), blk, 0, stream>>>(
      fmcat, c1_w, c1_b, fmc1, 786432, 262144);
  groupnorm_act<<<dim3(64, 16), dim3(256), 0, stream>>>(fmc1, c1_g, c1_be,
      4, 262144, 1);

  // fmp2[b,n,m] = sum_c fmc1[b,c,n] p2_w[m,c] + p2_b[m]   (M=n, N=m, K=c)
  gemm_wmma_f32<1, 1024, 1, 256, 1024, 1, 2><<<dim3(4, 32, 16), blk, 0, stream>>>(
      fmc1, p2_w, fmp2, 256, 262144, 0, 1048576, p2_b, 1.f);

  // attn[b,i,j] = softmax_j(fmp2[b,j,i] + fmp2[b,i,j])   (reuses attn buffer)
  transadd_softmax<<<dim3(1024, 16), dim3(256), 0, stream>>>(fmp2, attn);

  // outb[b,c,i] = sum_j attn[b,i,j] v[b,c,j]   (M=c(32), N=i, K=j, batch 16)
  gemm_wmma_f32<1024, 1, 1, 1024, 1024, 1, 0><<<dim3(4, 1, 16), blk, 0, stream>>>(
      vbuf, attn, outb, 1024, 32768, 1048576, 32768, nullptr, 1.f);

  // final projection: out[b,o,p] = wo[o,c] outb[b,c,p] + bo[o]   (batch 2)
  gemm_wmma_f32<256, 1, 1024, 1, 1024, 1, 1><<<dim3(4, 8, 2), blk, 0, stream>>>(
      wo, outb, out, 256, 0, 262144, 262144, bo, 1.f);
}